// SparseUpsampleBlock_7284264534487
// MI455X (gfx1250) — compile-verified
//
#include <hip/hip_runtime.h>

// ---------------- problem constants (match reference) ----------------
#define CIN   128
#define COUT  64
#define BDIM  2
#define TDIM  8
#define YODIM 255
#define XODIM 255

typedef __attribute__((ext_vector_type(2))) float v2f;
typedef __attribute__((ext_vector_type(8))) float v8f;

// ---------------- zero fill ----------------
__global__ void zero_kernel(float* __restrict__ p, long n) {
    long i = (long)blockIdx.x * blockDim.x + threadIdx.x;
    long stride = (long)gridDim.x * blockDim.x;
    for (; i < n; i += stride) p[i] = 0.0f;
}

// ---------------- GEMM (fp32 WMMA) + scatter-add ----------------
// One block = one tile of 16 points. 4 waves split the 27 kernel offsets.
// Per offset, a wave computes [16 x 64] = A[16x128] * W_k[128x64] with
// v_wmma_f32_16x16x4_f32 (32 K-steps x 4 N-tiles), then atomically
// scatters rows into the dense output grid.
__global__ __launch_bounds__(128) void spdeconv_gemm_scatter(
    const float* __restrict__ feats, const float* __restrict__ weight,
    const int* __restrict__ cb, const int* __restrict__ ct,
    const int* __restrict__ cy, const int* __restrict__ cx,
    float* __restrict__ out, unsigned int* __restrict__ cnt, int n_points)
{
    __shared__ float ldsA[16 * CIN];   // 8 KB A tile
    __shared__ int   ldsC[16 * 4];     // per-point coords (b,t,y,x)

    const int tid  = threadIdx.x;          // 0..127
    const int base = blockIdx.x * 16;

    // stage A tile (pad missing rows with zeros)
    for (int i = tid; i < 16 * CIN; i += 128) {
        int r = i / CIN;
        ldsA[i] = (base + r < n_points) ? feats[(long)(base + r) * CIN + (i % CIN)] : 0.0f;
    }
    // stage coords (sentinel for padded rows)
    if (tid < 16) {
        int p = base + tid;
        bool v = p < n_points;
        ldsC[tid * 4 + 0] = v ? cb[p] : -1000;
        ldsC[tid * 4 + 1] = v ? ct[p] : 0;
        ldsC[tid * 4 + 2] = v ? cy[p] : 0;
        ldsC[tid * 4 + 3] = v ? cx[p] : 0;
    }
    __syncthreads();

    const int wave = tid >> 5;
    const int lane = tid & 31;
    const int h    = lane >> 4;    // lane half
    const int n16  = lane & 15;    // N index within 16-wide tile / M row for A
    const int m    = lane & 15;    // A row (both halves map M=0..15)

    for (int o = wave; o < 27; o += 4) {          // wave-uniform loop
        const int kt = o / 9, ky = (o / 3) % 3, kx = o % 3;
        const float* __restrict__ W = weight + (long)o * CIN * COUT;

        v8f c0 = {}, c1 = {}, c2 = {}, c3 = {};

        for (int k0 = 0; k0 < CIN; k0 += 4) {
            const int kk = k0 + 2 * h;
            // A fragment: lane half h holds K = kk, kk+1 for row m
            v2f a;
            a.x = ldsA[m * CIN + kk];
            a.y = ldsA[m * CIN + kk + 1];
            // B fragments: lane half h holds rows K = kk, kk+1 at column n
            const float* wk = W + (long)kk * COUT + n16;
            v2f b0; b0.x = wk[0];  b0.y = wk[COUT];
            v2f b1; b1.x = wk[16]; b1.y = wk[COUT + 16];
            v2f b2; b2.x = wk[32]; b2.y = wk[COUT + 32];
            v2f b3; b3.x = wk[48]; b3.y = wk[COUT + 48];

            c0 = __builtin_amdgcn_wmma_f32_16x16x4_f32(false, a, false, b0, (short)0, c0, false, false);
            c1 = __builtin_amdgcn_wmma_f32_16x16x4_f32(false, a, false, b1, (short)0, c1, false, false);
            c2 = __builtin_amdgcn_wmma_f32_16x16x4_f32(false, a, false, b2, (short)0, c2, false, false);
            c3 = __builtin_amdgcn_wmma_f32_16x16x4_f32(false, a, false, b3, (short)0, c3, false, false);
        }

        // scatter: VGPR r of C/D holds row M = r + 8*h, column N = n16
        #pragma unroll
        for (int r = 0; r < 8; ++r) {
            const int mr = r + 8 * h;
            const int pb = ldsC[mr * 4 + 0];
            const int pt = ldsC[mr * 4 + 1];
            const int py = ldsC[mr * 4 + 2];
            const int px = ldsC[mr * 4 + 3];
            const int ot = pt - 1 + kt;
            const int oy = 2 * py - 1 + ky;
            const int ox = 2 * px - 1 + kx;
            const bool valid = (pb >= 0) &&
                               ((unsigned)ot < (unsigned)TDIM) &&
                               ((unsigned)oy < (unsigned)YODIM) &&
                               ((unsigned)ox < (unsigned)XODIM);
            if (valid) {
                const long site = (((long)pb * TDIM + ot) * YODIM + oy) * (long)XODIM + ox;
                float* dptr = out + site * COUT + n16;
                atomicAdd(dptr +  0, c0[r]);
                atomicAdd(dptr + 16, c1[r]);
                atomicAdd(dptr + 32, c2[r]);
                atomicAdd(dptr + 48, c3[r]);
                if (n16 == 0) cnt[site] = 1u;   // idempotent mask store
            }
        }
    }
}

// ---------------- per-channel sum / sumsq over the dense grid ----------------
// stride % 64 == 0 so each thread owns a fixed channel c = threadIdx.x & 63.
__global__ __launch_bounds__(256) void bn_stats(
    const float* __restrict__ out, float* __restrict__ sums,
    float* __restrict__ sumsq, long nelem)
{
    __shared__ float s1[256];
    __shared__ float s2[256];
    long i = (long)blockIdx.x * 256 + threadIdx.x;
    const long stride = (long)gridDim.x * 256;
    float s = 0.0f, q = 0.0f;
    for (; i < nelem; i += stride) {
        float v = out[i];
        s += v;
        q += v * v;
    }
    s1[threadIdx.x] = s;
    s2[threadIdx.x] = q;
    __syncthreads();
    if (threadIdx.x < 64) {
        float a = s1[threadIdx.x] + s1[threadIdx.x + 64] + s1[threadIdx.x + 128] + s1[threadIdx.x + 192];
        float b = s2[threadIdx.x] + s2[threadIdx.x + 64] + s2[threadIdx.x + 128] + s2[threadIdx.x + 192];
        atomicAdd(&sums[threadIdx.x], a);
        atomicAdd(&sumsq[threadIdx.x], b);
    }
}

// ---------------- active-site count ----------------
__global__ __launch_bounds__(256) void count_active(
    const unsigned int* __restrict__ cnt, float* __restrict__ nact, long nsites)
{
    __shared__ float s[256];
    long i = (long)blockIdx.x * 256 + threadIdx.x;
    const long stride = (long)gridDim.x * 256;
    float c = 0.0f;
    for (; i < nsites; i += stride) c += (cnt[i] != 0u) ? 1.0f : 0.0f;
    s[threadIdx.x] = c;
    __syncthreads();
    for (int off = 128; off > 0; off >>= 1) {
        if (threadIdx.x < off) s[threadIdx.x] += s[threadIdx.x + off];
        __syncthreads();
    }
    if (threadIdx.x == 0) atomicAdd(nact, s[0]);
}

// ---------------- normalize + ReLU + mask, in place ----------------
__global__ __launch_bounds__(256) void bn_apply(
    float* __restrict__ out, const unsigned int* __restrict__ cnt,
    const float* __restrict__ sums, const float* __restrict__ sumsq,
    const float* __restrict__ nact, const float* __restrict__ gamma,
    const float* __restrict__ beta, long nelem)
{
    const int c = threadIdx.x & 63;          // fixed channel (stride % 64 == 0)
    const float n = fmaxf(nact[0], 1.0f);
    const float mean = sums[c] / n;
    const float var  = sumsq[c] / n - mean * mean;
    const float rstd = rsqrtf(var + 1e-5f);
    const float g = gamma[c] * rstd;
    const float b = beta[c];
    long i = (long)blockIdx.x * 256 + threadIdx.x;
    const long stride = (long)gridDim.x * 256;
    for (; i < nelem; i += stride) {
        const float mask = (cnt[i >> 6] != 0u) ? 1.0f : 0.0f;
        const float v = (out[i] - mean) * g + b;
        out[i] = fmaxf(v, 0.0f) * mask;
    }
}

// ---------------- launch ----------------
extern "C" void kernel_launch(void* const* d_in, const int* in_sizes, int n_in,
                              void* d_out, int out_size, void* d_ws, size_t ws_size,
                              hipStream_t stream) {
    const float* feats  = (const float*)d_in[0];
    const float* weight = (const float*)d_in[1];
    const float* gamma  = (const float*)d_in[2];
    const float* beta   = (const float*)d_in[3];
    const int*   cb     = (const int*)d_in[4];
    const int*   ct     = (const int*)d_in[5];
    const int*   cy     = (const int*)d_in[6];
    const int*   cx     = (const int*)d_in[7];
    const int n_points  = in_sizes[4];

    const long nsites = (long)BDIM * TDIM * YODIM * XODIM;   // 1,040,400
    const long nelem  = nsites * COUT;                       // 66,585,600

    // workspace layout: [cnt grid (u32 x nsites)] [sums x64] [sumsq x64] [nact x1]
    const size_t acc_off = ((nsites * sizeof(unsigned int) + 255) / 256) * 256;
    unsigned int* cnt = (unsigned int*)d_ws;
    float* sums  = (float*)((char*)d_ws + acc_off);
    float* sumsq = sums + 64;
    float* nact  = sums + 128;
    float* out   = (float*)d_out;

    // 1. zero output grid + workspace (cnt + accumulators)
    zero_kernel<<<2048, 256, 0, stream>>>(out, nelem);
    zero_kernel<<<512, 256, 0, stream>>>((float*)d_ws, (long)(acc_off / 4 + 129));

    // 2. WMMA GEMM + scatter-add
    const int tiles = (n_points + 15) / 16;
    spdeconv_gemm_scatter<<<tiles, 128, 0, stream>>>(feats, weight, cb, ct, cy, cx,
                                                     out, cnt, n_points);

    // 3. BN statistics
    bn_stats<<<2048, 256, 0, stream>>>(out, sums, sumsq, nelem);
    count_active<<<1024, 256, 0, stream>>>(cnt, nact, nsites);

    // 4. normalize + ReLU + mask
    bn_apply<<<4096, 256, 0, stream>>>(out, cnt, sums, sumsq, nact, gamma, beta, nelem);
}